// YOLODetectionLoss_31086973288808
// MI455X (gfx1250) — compile-verified
//
#include <hip/hip_runtime.h>
#include <hip/hip_bf16.h>

#define NBLK   1024          // blocks per streaming kernel
#define NTHR   256           // threads per block (8 waves on wave32)
#define BN     1600          // B*N targets = 32*50
#define STRIDE 2624          // NBLK + BN, == 41*64 exactly
#define CHUNKS 41            // STRIDE / 64
#define CCH    85            // 5 + 80 channels

typedef __attribute__((ext_vector_type(2))) float v2f;
typedef __attribute__((ext_vector_type(8))) float v8f;

__device__ __forceinline__ float clip10(float x) {
    return fminf(fmaxf(x, -10.f), 10.f);
}
// numerically stable BCE-with-logits against target 0 (clipped logit)
__device__ __forceinline__ float bce0(float x) {
    return fmaxf(x, 0.f) + __logf(1.f + __expf(-fabsf(x)));
}

// -------- Stage 1: dense streaming pass (target == 0 everywhere) --------
__global__ void yolo_stream(const float4* __restrict__ pred, int total4, int log2hw4,
                            float* __restrict__ part /* = W + s*3*STRIDE */) {
    float accBox = 0.f, accObj = 0.f, accCls = 0.f;
    const int tid  = blockIdx.x * blockDim.x + threadIdx.x;
    const int nthr = gridDim.x * blockDim.x;
    for (int i = tid; i < total4; i += nthr) {
        if (i + nthr < total4)
            __builtin_prefetch(&pred[i + nthr], 0, 0);   // global_prefetch on gfx1250
        float4 v = pred[i];
        int c = (i >> log2hw4) % CCH;                    // channel (wave-uniform almost always)
        float x0 = clip10(v.x), x1 = clip10(v.y), x2 = clip10(v.z), x3 = clip10(v.w);
        if (c < 4) {
            accBox += x0*x0 + x1*x1 + x2*x2 + x3*x3;     // MSE vs 0
        } else {
            float bs = bce0(x0) + bce0(x1) + bce0(x2) + bce0(x3);
            if (c == 4) accObj += bs; else accCls += bs;
        }
    }
    // deterministic block reduction (fixed LDS tree order)
    __shared__ float sm[NTHR];
    float acc[3] = {accBox, accObj, accCls};
    float res[3];
    for (int t = 0; t < 3; ++t) {
        sm[threadIdx.x] = acc[t];
        __syncthreads();
        for (int off = NTHR / 2; off > 0; off >>= 1) {
            if (threadIdx.x < off) sm[threadIdx.x] += sm[threadIdx.x + off];
            __syncthreads();
        }
        res[t] = sm[0];
        __syncthreads();
    }
    if (threadIdx.x == 0) {
        part[0 * STRIDE + blockIdx.x] = res[0];
        part[1 * STRIDE + blockIdx.x] = res[1];
        part[2 * STRIDE + blockIdx.x] = res[2];
    }
}

// -------- Stage 2: sparse corrections at the 1600 scattered target cells --------
__global__ void yolo_corrections(const float* __restrict__ pred,
                                 const float* __restrict__ boxes,
                                 const int*   __restrict__ labels,
                                 int h, int w,
                                 float* __restrict__ part /* = W + s*3*STRIDE */) {
    int j = blockIdx.x * blockDim.x + threadIdx.x;
    if (j >= BN) return;
    const int N = 50;
    int b = j / N, n = j % N;
    const float* bx = boxes + (size_t)(b * N + n) * 4;
    float gx = bx[0] * (float)w, gy = bx[1] * (float)h;
    int gi = min((int)gx, w - 1);
    int gj = min((int)gy, h - 1);
    int lab = labels[b * N + n];

    // scatter .set semantics: last write wins -> this target counts only if no
    // later target (same image) hits the same cell (box/obj) / same cell+label (cls)
    bool lastCell = true, lastCellLabel = true;
    for (int m = n + 1; m < N; ++m) {
        const float* bm = boxes + (size_t)(b * N + m) * 4;
        int gi2 = min((int)(bm[0] * (float)w), w - 1);
        int gj2 = min((int)(bm[1] * (float)h), h - 1);
        if (gi2 == gi && gj2 == gj) {
            lastCell = false;
            if (labels[b * N + m] == lab) lastCellLabel = false;
        }
    }

    const size_t hw   = (size_t)h * w;
    const size_t base = (size_t)b * CCH * hw + (size_t)gj * w + gi;

    float boxCorr = 0.f, objCorr = 0.f, clsCorr = 0.f;
    if (lastCell) {
        float vals[4] = {gx - (float)gi, gy - (float)gj, bx[2], bx[3]};
        #pragma unroll
        for (int ch = 0; ch < 4; ++ch) {
            float p = clip10(pred[base + (size_t)ch * hw]);
            float v = vals[ch];
            boxCorr += v * v - 2.f * p * v;   // (p-v)^2 - p^2
        }
        objCorr = -clip10(pred[base + 4 * hw]);        // BCE(x,1)-BCE(x,0) = -x
    }
    if (lastCellLabel) {
        clsCorr = -clip10(pred[base + (size_t)(5 + lab) * hw]);
    }
    part[0 * STRIDE + NBLK + j] = boxCorr;
    part[1 * STRIDE + NBLK + j] = objCorr;
    part[2 * STRIDE + NBLK + j] = clsCorr;
}

// -------- Stage 3: WMMA-based deterministic reduction + scalar finish --------
// One wave. For each of the 9 (scale,term) arrays (2624 floats), chain
// D = ones(16x4) * B(4x16) + D : every V_WMMA_F32_16X16X4_F32 folds 64 fresh
// values into the accumulator. Row 0 of D then holds 16 column sums in
// lanes 0..15 of the first D VGPR; 4 shfl_xor steps finish the sum.
__global__ void yolo_finalize(const float* __restrict__ W, float* __restrict__ out) {
    const int lane = threadIdx.x;          // block is exactly 32 threads, EXEC all ones
    __shared__ float terms[9];

    v2f a;
    a[0] = 1.f; a[1] = 1.f;                // A = all-ones 16x4

    for (int st = 0; st < 9; ++st) {
        const float* base = W + (size_t)st * STRIDE;
        v8f c = {};
        for (int k = 0; k < CHUNKS; ++k) {
            v2f bb;
            bb[0] = base[k * 64 + lane];
            bb[1] = base[k * 64 + 32 + lane];
            c = __builtin_amdgcn_wmma_f32_16x16x4_f32(
                    /*neg_a=*/false, a, /*neg_b=*/false, bb,
                    /*c_mod=*/(short)0, c, /*reuse_a=*/false, /*reuse_b=*/false);
        }
        float v = c[0];                                   // M=0 row: N=0..15 in lanes 0..15
        #pragma unroll
        for (int off = 8; off > 0; off >>= 1)
            v += __shfl_xor(v, off, 32);                  // sum lanes 0..15
        if (lane == 0) terms[st] = v;
    }
    __syncthreads();

    if (lane == 0) {
        const float hws[3] = {4096.f, 1024.f, 256.f};
        float total = 0.f;
        for (int s = 0; s < 3; ++s) {
            float denom = 32.f * hws[s];                  // b*h*w
            float box = terms[s * 3 + 0] / (denom * 4.f);
            float obj = terms[s * 3 + 1] / denom;
            float cls = terms[s * 3 + 2] / (denom * 80.f);
            box = isfinite(box) ? box : 0.f;
            obj = isfinite(obj) ? obj : 0.f;
            cls = isfinite(cls) ? cls : 0.f;
            total += 5.f * box + 1.f * obj + 1.f * cls;
        }
        total = total / 3.f;
        out[0] = fminf(fmaxf(total, 0.f), 1000000.f);
    }
}

extern "C" void kernel_launch(void* const* d_in, const int* in_sizes, int n_in,
                              void* d_out, int out_size, void* d_ws, size_t ws_size,
                              hipStream_t stream) {
    (void)in_sizes; (void)n_in; (void)out_size; (void)ws_size;
    const float* preds[3] = {(const float*)d_in[0], (const float*)d_in[1], (const float*)d_in[2]};
    const float* boxes    = (const float*)d_in[3];
    const int*   labels   = (const int*)d_in[4];
    float* W   = (float*)d_ws;     // 9 * STRIDE floats = ~94 KB, fully rewritten each call
    float* out = (float*)d_out;

    const int hw[3]      = {4096, 1024, 256};
    const int log2hw4[3] = {10, 8, 6};      // log2(hw/4)
    const int hdim[3]    = {64, 32, 16};

    for (int s = 0; s < 3; ++s) {
        int total4 = 32 * CCH * hw[s] / 4;
        yolo_stream<<<NBLK, NTHR, 0, stream>>>(
            (const float4*)preds[s], total4, log2hw4[s], W + (size_t)s * 3 * STRIDE);
        yolo_corrections<<<(BN + NTHR - 1) / NTHR, NTHR, 0, stream>>>(
            preds[s], boxes, labels, hdim[s], hdim[s], W + (size_t)s * 3 * STRIDE);
    }
    yolo_finalize<<<1, 32, 0, stream>>>(W, out);
}